// PrototypeMetaLinearUnifiedModel_68994354643460
// MI455X (gfx1250) — compile-verified
//
#include <hip/hip_runtime.h>
#include <hip/hip_bf16.h>

// ---------------------------------------------------------------------------
// PrototypeMetaLinear fused pipeline for MI455X (gfx1250, wave32)
//
//   proto_c = mean(x[labels==c])            (pass 1: atomic column reduction)
//   d_c     = sqrt(|x|^2 - 2 x.proto_c + |proto_c|^2)
//   out     = (x W1^T + b1)/d1 + (x W2^T + b2)/d2
//
// Pass 2 computes 6 simultaneous length-768 dot products per row
// (p1, p2, W1 rows, W2 rows) via V_WMMA_F32_16X16X4_F32, RHS packed as
// Bmat[768][16] (cols 6..15 zero, hot in WGP$). Memory-bound: two passes
// over x (~0.8 GB -> ~35 us at 23.3 TB/s), so x moves at b128/b96
// granularity; the x tile goes global->LDS via GLOBAL_LOAD_ASYNC_TO_LDS_B128
// (ASYNCcnt) when the toolchain exposes the builtin.
// ---------------------------------------------------------------------------

typedef float v2f __attribute__((ext_vector_type(2)));
typedef float v8f __attribute__((ext_vector_type(8)));
// builtin_amdgcn_global_load_async_to_lds_b128 takes pointers to GCC-style
// <4 x i32> vectors in AS1 (global) / AS3 (LDS) — per hipcc diagnostic.
typedef int vs4i __attribute__((vector_size(16)));
typedef __attribute__((address_space(1))) vs4i GAS4;
typedef __attribute__((address_space(3))) vs4i LAS4;

#if defined(__has_builtin)
#  if __has_builtin(__builtin_amdgcn_global_load_async_to_lds_b128)
#    define HAVE_ASYNC_LDS 1
#  endif
#endif
#ifndef HAVE_ASYNC_LDS
#  define HAVE_ASYNC_LDS 0
#endif

#if defined(__has_builtin)
#  if __has_builtin(__builtin_amdgcn_s_wait_asynccnt)
#    define WAIT_ASYNC0() __builtin_amdgcn_s_wait_asynccnt(0)
#  endif
#endif
#ifndef WAIT_ASYNC0
#  define WAIT_ASYNC0() asm volatile("s_wait_asynccnt 0" ::: "memory")
#endif

#define N_ROWS 131072
#define D_DIM  768
#define TPAD   36        // tile row stride: 144B -> 16B aligned rows, and
                         // 36*dr % 64 never equals 2 -> halves never collide

// workspace layout (floats)
#define WS_S1    0       // 768: sum of rows with label==p1
#define WS_S2    768     // 768: sum of rows with label==p2
#define WS_CNT1  1536
#define WS_CNT2  1537
#define WS_PN1   1538    // |proto1|^2
#define WS_PN2   1539    // |proto2|^2
#define WS_ZERO_END 1540
#define WS_BMAT  1552    // 768*16: packed B matrix, row-major [k][16] (16B aligned)

// ---------------------------------------------------------------------------
__global__ __launch_bounds__(256) void proto_init(float* __restrict__ ws) {
    int i = blockIdx.x * 256 + threadIdx.x;
    if (i < WS_ZERO_END) ws[i] = 0.0f;
}

// ---------------------------------------------------------------------------
// Pass 1: class sums + counts. 512 blocks x 256 threads; block handles 256
// rows, thread t owns contiguous columns {3t, 3t+1, 3t+2} so the three row
// loads fuse into one global_load_b96 (coalesced 3KB/row across the block).
__global__ __launch_bounds__(256) void proto_classsum(
    const float* __restrict__ x, const int* __restrict__ labels,
    const int* __restrict__ p1p, const int* __restrict__ p2p,
    float* __restrict__ ws)
{
    __shared__ int   slab[256];
    __shared__ float scnt[2];
    const int tid = threadIdx.x;
    const int col = tid * 3;
    const int rowBase = blockIdx.x * 256;
    const int p1 = p1p[0], p2 = p2p[0];

    if (tid < 2) scnt[tid] = 0.0f;
    const int myLab = labels[rowBase + tid];
    slab[tid] = myLab;
    __syncthreads();
    atomicAdd(&scnt[0], (myLab == p1) ? 1.0f : 0.0f);
    atomicAdd(&scnt[1], (myLab == p2) ? 1.0f : 0.0f);

    float a10 = 0.f, a11 = 0.f, a12 = 0.f;
    float a20 = 0.f, a21 = 0.f, a22 = 0.f;
    for (int r = 0; r < 256; ++r) {
        const int lab = slab[r];
        const float* xr = x + (size_t)(rowBase + r) * D_DIM + col;
        const float v0 = xr[0];
        const float v1 = xr[1];
        const float v2 = xr[2];
        if (lab == p1) { a10 += v0; a11 += v1; a12 += v2; }
        if (lab == p2) { a20 += v0; a21 += v1; a22 += v2; }
    }
    atomicAdd(&ws[WS_S1 + col    ], a10);
    atomicAdd(&ws[WS_S1 + col + 1], a11);
    atomicAdd(&ws[WS_S1 + col + 2], a12);
    atomicAdd(&ws[WS_S2 + col    ], a20);
    atomicAdd(&ws[WS_S2 + col + 1], a21);
    atomicAdd(&ws[WS_S2 + col + 2], a22);
    __syncthreads();
    if (tid == 0) {
        atomicAdd(&ws[WS_CNT1], scnt[0]);
        atomicAdd(&ws[WS_CNT2], scnt[1]);
    }
}

// ---------------------------------------------------------------------------
// Finalize: protos = sums / max(cnt,1); pack Bmat[k][0..15] =
// {p1, p2, W1r0, W1r1, W2r0, W2r1, 0 x 10}; block-reduce |p|^2.
__global__ __launch_bounds__(256) void proto_finalize(
    const float* __restrict__ W1, const float* __restrict__ W2,
    float* __restrict__ ws)
{
    __shared__ float red1[256], red2[256];
    const int tid = threadIdx.x;
    const float i1 = 1.0f / fmaxf(ws[WS_CNT1], 1.0f);
    const float i2 = 1.0f / fmaxf(ws[WS_CNT2], 1.0f);
    float p1sq = 0.f, p2sq = 0.f;
    for (int k = tid; k < D_DIM; k += 256) {
        const float p1k = ws[WS_S1 + k] * i1;
        const float p2k = ws[WS_S2 + k] * i2;
        p1sq += p1k * p1k;
        p2sq += p2k * p2k;
        float* bm = ws + WS_BMAT + (size_t)k * 16;
        bm[0] = p1k;              bm[1] = p2k;
        bm[2] = W1[k];            bm[3] = W1[D_DIM + k];
        bm[4] = W2[k];            bm[5] = W2[D_DIM + k];
        #pragma unroll
        for (int j = 6; j < 16; ++j) bm[j] = 0.0f;
    }
    red1[tid] = p1sq; red2[tid] = p2sq;
    __syncthreads();
    for (int s = 128; s > 0; s >>= 1) {
        if (tid < s) { red1[tid] += red1[tid + s]; red2[tid] += red2[tid + s]; }
        __syncthreads();
    }
    if (tid == 0) { ws[WS_PN1] = red1[0]; ws[WS_PN2] = red2[0]; }
}

// ---------------------------------------------------------------------------
// Pass 2 (main): 256 threads = 8 waves, each wave owns 16 rows.
// K loop in chunks of 32 staged through LDS at b128 granularity
// (async global->LDS when available), 8 WMMA k-steps of 4.
//
// A 16x4 f32 fragment (ISA 7.12.2): lane l<16 holds row l, K={k, k+1};
// lane l>=16 holds row l-16, K={k+2, k+3}. B 4x16 mirrored: lane l<16 holds
// N=l, K={k, k+1}; lane l>=16 holds N=l-16, K={k+2, k+3}.
// C/D 16x16 f32: VGPR g, lane l -> (M = g + (l<16?0:8), N = l%16).
__global__ __launch_bounds__(256) void proto_main(
    const float* __restrict__ x, const float* __restrict__ b1,
    const float* __restrict__ b2, const float* __restrict__ ws,
    float* __restrict__ out)
{
    __shared__ __align__(16) float tile[128][TPAD];  // 128x32 payload, 18KB
    __shared__ float cstage[8][16][8];               // per-wave C gather staging

    const int tid  = threadIdx.x;
    const int lane = tid & 31;
    const int w    = tid >> 5;
    const int n16  = lane & 15;
    const int half = (lane < 16) ? 0 : 2;  // K sub-offset for this half-wave
    const int rowBlock = blockIdx.x * 128;
    const int rowLocal = w * 16 + n16;     // tile row this lane's A covers
    const float* __restrict__ bmat = ws + WS_BMAT;

    // this thread's 4 float4 slots of the 128x32 tile (1024 float4 total)
    const int f0 = tid;                    // float4 index base, +256 per i
    v8f   acc = {};
    float nsq = 0.0f;                      // partial |x|^2 (half the K values)

    for (int kc = 0; kc < D_DIM; kc += 32) {
        __syncthreads();                   // all waves done reading prev tile
#if HAVE_ASYNC_LDS
        #pragma unroll
        for (int i = 0; i < 4; ++i) {
            const int f = f0 + i * 256;    // float4 index
            const int r = f >> 3;          // 8 float4 per 32-col row
            const int c = (f & 7) * 4;
            __builtin_amdgcn_global_load_async_to_lds_b128(
                (GAS4*)(x + (size_t)(rowBlock + r) * D_DIM + kc + c),
                (LAS4*)&tile[r][c], 0, 0);
        }
        WAIT_ASYNC0();                     // LDS writes landed (this wave)
#else
        #pragma unroll
        for (int i = 0; i < 4; ++i) {
            const int f = f0 + i * 256;
            const int r = f >> 3;
            const int c = (f & 7) * 4;
            const float4 v = *(const float4*)(x + (size_t)(rowBlock + r) * D_DIM + kc + c);
            *(float4*)&tile[r][c] = v;
        }
#endif
        __syncthreads();                   // publish tile to all waves
        #pragma unroll
        for (int s = 0; s < 8; ++s) {
            const int kl = s * 4 + half;
            v2f a, b;
            a[0] = tile[rowLocal][kl];
            a[1] = tile[rowLocal][kl + 1];
            const int kg = kc + kl;
            b[0] = bmat[(size_t)kg * 16 + n16];
            b[1] = bmat[(size_t)(kg + 1) * 16 + n16];
            nsq += a[0] * a[0] + a[1] * a[1];
            acc = __builtin_amdgcn_wmma_f32_16x16x4_f32(
                false, a, false, b, (short)0, acc, false, false);
        }
    }

    // |x_row|^2 = this half-wave partial + opposite half-wave partial
    const float nOther = __shfl(nsq, n16 + 16, 32);
    const float nx2 = nsq + nOther;        // valid in lanes 0..15

    // stage the 6 useful C columns: lane holds N=n16 for M = g + (l<16?0:8)
    if (n16 < 6) {
        const int mbase = (lane < 16) ? 0 : 8;
        #pragma unroll
        for (int g = 0; g < 8; ++g)
            cstage[w][mbase + g][n16] = acc[g];
    }
    __syncthreads();

    if (lane < 16) {
        const int row = rowBlock + w * 16 + lane;
        const float q1  = cstage[w][lane][0];
        const float q2  = cstage[w][lane][1];
        const float y10 = cstage[w][lane][2];
        const float y11 = cstage[w][lane][3];
        const float y20 = cstage[w][lane][4];
        const float y21 = cstage[w][lane][5];
        const float pn1 = ws[WS_PN1], pn2 = ws[WS_PN2];
        const float d1 = sqrtf(fmaxf(nx2 - 2.0f * q1 + pn1, 0.0f));
        const float d2 = sqrtf(fmaxf(nx2 - 2.0f * q2 + pn2, 0.0f));
        const float r1 = 1.0f / d1, r2 = 1.0f / d2;
        const float b10 = b1[0], b11 = b1[1];
        const float b20 = b2[0], b21 = b2[1];
        out[(size_t)row * 2 + 0] = (y10 + b10) * r1 + (y20 + b20) * r2;
        out[(size_t)row * 2 + 1] = (y11 + b11) * r1 + (y21 + b21) * r2;
        out[(size_t)2 * N_ROWS + row] = d1;   // d1 block
        out[(size_t)3 * N_ROWS + row] = d2;   // d2 block
    }
}

// ---------------------------------------------------------------------------
extern "C" void kernel_launch(void* const* d_in, const int* in_sizes, int n_in,
                              void* d_out, int out_size, void* d_ws, size_t ws_size,
                              hipStream_t stream) {
    const float* x      = (const float*)d_in[0];
    const int*   labels = (const int*)  d_in[1];
    const float* W1     = (const float*)d_in[2];
    const float* b1     = (const float*)d_in[3];
    const float* W2     = (const float*)d_in[4];
    const float* b2     = (const float*)d_in[5];
    const int*   p1     = (const int*)  d_in[6];
    const int*   p2     = (const int*)  d_in[7];
    float* ws  = (float*)d_ws;
    float* out = (float*)d_out;

    proto_init     <<<(WS_ZERO_END + 255) / 256, 256, 0, stream>>>(ws);
    proto_classsum <<<N_ROWS / 256, 256, 0, stream>>>(x, labels, p1, p2, ws);
    proto_finalize <<<1, 256, 0, stream>>>(W1, W2, ws);
    proto_main     <<<N_ROWS / 128, 256, 0, stream>>>(x, b1, b2, ws, out);
}